// WaveFoward_51883204935988
// MI455X (gfx1250) — compile-verified
//
#include <hip/hip_runtime.h>
#include <hip/hip_bf16.h>

// ---------------------------------------------------------------------------
// Wave forward model on gfx1250 (MI455X).
// 16 independent 512x512 f32 wavefields, 48 steps, 4th-order periodic
// Laplacian done with V_WMMA_F32_16X16X4_F32 (stencil-as-banded-matmul):
//   d2_vert tile = S(16x20) @ P(20x16)   -> 5 WMMAs (K=4 chunks)
//   d2_horz tile = P(16x20) @ S^T(20x16) -> 5 WMMAs (K=4 chunks)
// accumulated into one 16x16 f32 accumulator. Working set (~35 MB) is
// L2-resident on the 192 MB L2, so each step is a single lean kernel.
// ---------------------------------------------------------------------------

#define NXG   512
#define NXMSK 511
#define FIELD (512 * 512)
#define NR    16
#define NM    128
#define NT    48
#define ABCP  55            // ABC_PD = 512/2 - 200 - 1
#define ILO   128           // ABC_PD + PD = 55 + 73
#define IHI   384           // ILO + 256
#define EHI   457           // 512 - ABC_PD
#define SRCC  0.09f         // (bg*dt/dx)^2 = (1.5*0.2)^2
#define BGV   1.5f
#define ABCV  1e-6f

typedef float v2f __attribute__((ext_vector_type(2)));
typedef float v8f __attribute__((ext_vector_type(8)));

// 4th-order second-derivative weights /12: w[0..4] = {-1,16,-30,16,-1}/12
__device__ __forceinline__ float wcoef(int d) {
    if (d < 0 || d > 4) return 0.0f;
    if (d == 2) return -30.0f / 12.0f;
    if (d == 1 || d == 3) return 16.0f / 12.0f;
    return -1.0f / 12.0f;
}

__global__ __launch_bounds__(256) void wf_init_fields(float* b0, float* b1) {
    int i = blockIdx.x * 256 + threadIdx.x;
    b0[i] = 0.0f;
    b1[i] = 0.0f;
}

__global__ __launch_bounds__(256) void wf_init_coef_map(const float* __restrict__ x,
                                                        float* __restrict__ coef,
                                                        short* __restrict__ map) {
    int i = blockIdx.x * 256 + threadIdx.x;       // 0..FIELD-1
    int row = i >> 9;
    int col = i & NXMSK;
    float v;
    if (row < ABCP || row >= EHI || col < ABCP || col >= EHI) {
        v = ABCV;
    } else if (row >= ILO && row < IHI && col >= ILO && col < IHI) {
        v = x[(row - ILO) * 256 + (col - ILO)];
    } else {
        v = BGV;
    }
    float dv = 0.2f * v;                          // dt*xp/dx
    coef[i] = dv * dv;
    map[i] = (short)-1;
}

__global__ void wf_set_map(const int* __restrict__ meas, short* __restrict__ map) {
    int m = threadIdx.x;                          // 128 threads
    map[meas[m]] = (short)m;
}

// One timestep: p_new = 2*p1 - p0 + coef*lap4(p1), + source, + measurement.
// p0buf is both the old p0 (read only at center, by its owning thread) and
// the destination, so two buffers ping-pong across steps.
__global__ __launch_bounds__(256) void wf_step(const float* __restrict__ p1,
                                               float* p0buf,
                                               const float* __restrict__ coef,
                                               const short* __restrict__ map,
                                               const int* __restrict__ trans,
                                               const float* __restrict__ srcI,
                                               int t,
                                               float* __restrict__ out) {
    __shared__ float lds[8][20 * 20];             // one 20x20 halo tile per wave

    const int wave = threadIdx.x >> 5;
    const int lane = threadIdx.x & 31;
    const int m    = lane & 15;                   // N (or M) position within tile
    const int hi   = lane >> 4;                   // 0: lanes 0-15, 1: lanes 16-31
    const int klo  = hi << 1;                     // K sub-offset per ISA A/B layout

    const int g    = blockIdx.x * 8 + wave;       // global tile id, 16*1024 total
    const int r    = g >> 10;                     // field 0..15
    const int tile = g & 1023;
    const int X0   = (tile >> 5) << 4;
    const int J0   = (tile & 31) << 4;

    const int    base = r * FIELD;
    const float* p1f  = p1 + base;
    float*       tp   = lds[wave];

    // Hint the epilogue streams into cache early.
    __builtin_prefetch(&p0buf[base + X0 * NXG + J0], 0, 1);
    __builtin_prefetch(&coef[X0 * NXG + J0], 0, 1);

    // Stage 20x20 halo tile of p1 with periodic wrap (roll == &511).
    for (int idx = lane; idx < 400; idx += 32) {
        int rr = idx / 20;
        int cc = idx - rr * 20;
        int gx = (X0 - 2 + rr) & NXMSK;
        int gy = (J0 - 2 + cc) & NXMSK;
        tp[idx] = p1f[gx * NXG + gy];
    }
    // Wave-private LDS slice: DS ops are in-order per wave; compiler inserts
    // the s_wait_dscnt before dependent reads below.

    // Constant stencil fragments: value w[K - M] (vertical A) == w[K - N]
    // (horizontal B) in the ISA wave32 16x4 / 4x16 f32 layouts.
    float cw0[5], cw1[5];
#pragma unroll
    for (int c = 0; c < 5; ++c) {
        int k = 4 * c + klo;
        cw0[c] = wcoef(k - m);
        cw1[c] = wcoef(k + 1 - m);
    }

    v8f acc = {};

    // Vertical derivative: acc += S @ P   (A const, B = tile rows)
#pragma unroll
    for (int c = 0; c < 5; ++c) {
        int kr = 4 * c + klo;
        v2f a; a.x = cw0[c];                  a.y = cw1[c];
        v2f b; b.x = tp[kr * 20 + m + 2];     b.y = tp[(kr + 1) * 20 + m + 2];
        acc = __builtin_amdgcn_wmma_f32_16x16x4_f32(false, a, false, b,
                                                    (short)0, acc, false, false);
    }
    // Horizontal derivative: acc += P @ S^T (A = tile cols, B const)
#pragma unroll
    for (int c = 0; c < 5; ++c) {
        int kc = 4 * c + klo;
        v2f a; a.x = tp[(m + 2) * 20 + kc];   a.y = tp[(m + 2) * 20 + kc + 1];
        v2f b; b.x = cw0[c];                  b.y = cw1[c];
        acc = __builtin_amdgcn_wmma_f32_16x16x4_f32(false, a, false, b,
                                                    (short)0, acc, false, false);
    }

    // Epilogue: leapfrog update + per-field source + measurement gather.
    const int   transFlat = trans[r];
    const float sAmp      = SRCC * srcI[t];
#pragma unroll
    for (int v = 0; v < 8; ++v) {
        int M    = v + (hi << 3);             // C/D layout: VGPR v, lane half
        int X    = X0 + M;
        int J    = J0 + m;
        int flat = X * NXG + J;
        float p1c = tp[(M + 2) * 20 + m + 2];
        float val = 2.0f * p1c - p0buf[base + flat] + coef[flat] * acc[v];
        if (flat == transFlat) val += sAmp;
        p0buf[base + flat] = val;
        int mi = map[flat];
        if (mi >= 0) out[(r * NM + mi) * NT + t] = val;
    }
}

extern "C" void kernel_launch(void* const* d_in, const int* in_sizes, int n_in,
                              void* d_out, int out_size, void* d_ws, size_t ws_size,
                              hipStream_t stream) {
    const float* x     = (const float*)d_in[0];   // (1,1,256,256) f32
    const float* srcI  = (const float*)d_in[1];   // (48,) f32
    const int*   meas  = (const int*)d_in[2];     // (128,) i32
    const int*   trans = (const int*)d_in[3];     // (16,) i32
    float*       out   = (float*)d_out;           // (1,16,128,48) f32

    float* buf0 = (float*)d_ws;
    float* buf1 = buf0 + (size_t)NR * FIELD;
    float* coef = buf1 + (size_t)NR * FIELD;
    short* map  = (short*)(coef + FIELD);
    // ws usage: 2*16*1MB + 1MB + 0.5MB ~= 35.1 MB

    wf_init_fields<<<(NR * FIELD) / 256, 256, 0, stream>>>(buf0, buf1);
    wf_init_coef_map<<<FIELD / 256, 256, 0, stream>>>(x, coef, map);
    wf_set_map<<<1, NM, 0, stream>>>(meas, map);

    for (int t = 0; t < NT; ++t) {
        const float* p1b = (t & 1) ? buf0 : buf1;  // t=0: p1 = zeros(buf1)
        float*       p0b = (t & 1) ? buf1 : buf0;  // old p0 / new p output
        // 16 fields * 32*32 tiles = 16384 waves; 8 waves (one tile each) / block
        wf_step<<<2048, 256, 0, stream>>>(p1b, p0b, coef, map, trans, srcI, t, out);
    }
}